// MultiHeadAttention_82746839925257
// MI455X (gfx1250) — compile-verified
//
#include <hip/hip_runtime.h>
#include <hip/hip_bf16.h>

// Problem constants
#define B_  16
#define S_  512
#define DM_ 256
#define H_  8
#define DK_ 32
#define DV_ 32
#define EPS_ 1e-5f

typedef float v2f __attribute__((ext_vector_type(2)));
typedef float v8f __attribute__((ext_vector_type(8)));

static __device__ __forceinline__ v8f wmma4(v2f a, v2f b, v8f c) {
  // D = A(16x4 f32) * B(4x16 f32) + C(16x16 f32)
  return __builtin_amdgcn_wmma_f32_16x16x4_f32(
      /*neg_a=*/false, a, /*neg_b=*/false, b,
      /*c_mod=*/(short)0, c, /*reuse_a=*/false, /*reuse_b=*/false);
}

// Async global->LDS copy, 16 bytes per lane (GLOBAL_LOAD_ASYNC_TO_LDS_B128,
// tracked by ASYNCcnt). LDS operand is the raw LDS offset = low 32 bits of
// the generic pointer.
static __device__ __forceinline__ void async_g2l_b128(void* lds, const void* g) {
  asm volatile("global_load_async_to_lds_b128 %0, %1, off"
               :: "v"((unsigned)(uintptr_t)lds), "v"(g) : "memory");
}
static __device__ __forceinline__ void wait_async0() {
  asm volatile("s_wait_asynccnt 0x0" ::: "memory");
}

// ---------------------------------------------------------------------------
// Generic tiled WMMA-f32 GEMM: C = A[M,K] @ W[K,N]
// Block = 128 threads (4 waves). WG tile = 64(M) x 32(N). K staged in LDS by 32.
// A tile copied with async global->LDS (no VGPR round-trip); W tile staged
// TRANSPOSED (Wt[n][k], stride 34) so every WMMA B-fragment is one aligned
// ds_load_b64.
// outMode: 0 = C[m*ldc+n]; 1 = head-split [B,H,S,32]; 2 = head-split
// transposed [B,H,32,S] (used for V so attention B-fragments are contiguous).
// ---------------------------------------------------------------------------
__global__ __launch_bounds__(128) void gemm_wmma_f32(
    const float* __restrict__ A, const float* __restrict__ W, float* __restrict__ C,
    int K, int lda, int ldw, int ldc,
    long strideA, long strideW, long strideC, int outMode)
{
  __shared__ float As[64 * 32];
  __shared__ float Wt[32 * 34];       // [n][k], padded row stride 34
  const int tid  = threadIdx.x;
  const int wave = tid >> 5;
  const int lane = tid & 31;
  const int half = lane >> 4;
  const int l16  = lane & 15;
  const int m0 = blockIdx.x * 64;
  const int n0 = blockIdx.y * 32;
  A += (long)blockIdx.z * strideA;
  W += (long)blockIdx.z * strideW;
  C += (long)blockIdx.z * strideC;

  v8f c0 = {}; v8f c1 = {};

  for (int kc = 0; kc < K; kc += 32) {
    // ---- A tile: 64x32 floats, async global->LDS, 4x b128 per thread ----
    #pragma unroll
    for (int r = 0; r < 4; ++r) {
      int fi  = tid + 128 * r;                    // float4 index
      int row = fi >> 3, c4 = fi & 7;
      async_g2l_b128(&As[row * 32 + c4 * 4],
                     &A[(long)(m0 + row) * lda + kc + c4 * 4]);
    }
    // ---- W tile: 32x32, load to regs then store transposed ----
    {
      int r0 = tid >> 3,          c40 = tid & 7;
      int r1 = (tid + 128) >> 3,  c41 = tid & 7;
      float4 wv0 = *(const float4*)&W[(long)(kc + r0) * ldw + n0 + c40 * 4];
      float4 wv1 = *(const float4*)&W[(long)(kc + r1) * ldw + n0 + c41 * 4];
      Wt[(c40 * 4 + 0) * 34 + r0] = wv0.x;
      Wt[(c40 * 4 + 1) * 34 + r0] = wv0.y;
      Wt[(c40 * 4 + 2) * 34 + r0] = wv0.z;
      Wt[(c40 * 4 + 3) * 34 + r0] = wv0.w;
      Wt[(c41 * 4 + 0) * 34 + r1] = wv1.x;
      Wt[(c41 * 4 + 1) * 34 + r1] = wv1.y;
      Wt[(c41 * 4 + 2) * 34 + r1] = wv1.z;
      Wt[(c41 * 4 + 3) * 34 + r1] = wv1.w;
    }
    if (kc + 32 < K)                              // global_prefetch_b8 next A chunk
      __builtin_prefetch(&A[(long)(m0 + (tid >> 1)) * lda + kc + 32], 0, 0);
    wait_async0();
    __syncthreads();

    const int am = wave * 16 + l16;
    #pragma unroll
    for (int ks = 0; ks < 8; ++ks) {
      int kb = ks * 4 + 2 * half;
      v2f a  = *(const v2f*)&As[am * 32 + kb];
      v2f b0 = *(const v2f*)&Wt[l16 * 34 + kb];          // aligned b64
      v2f b1 = *(const v2f*)&Wt[(16 + l16) * 34 + kb];
      c0 = wmma4(a, b0, c0);
      c1 = wmma4(a, b1, c1);
    }
    __syncthreads();
  }

  const int gmBase = m0 + wave * 16;
  #pragma unroll
  for (int r = 0; r < 8; ++r) {
    int row = r + 8 * half;
    int gm  = gmBase + row;
    #pragma unroll
    for (int nt = 0; nt < 2; ++nt) {
      int gc  = n0 + nt * 16 + l16;
      float v = (nt == 0) ? c0[r] : c1[r];
      if (outMode == 1) {
        int bb = gm >> 9, s = gm & 511;
        int h  = gc >> 5, d = gc & 31;
        C[(((long)bb * H_ + h) * S_ + s) * 32 + d] = v;
      } else if (outMode == 2) {
        int bb = gm >> 9, s = gm & 511;
        int h  = gc >> 5, d = gc & 31;
        C[(((long)bb * H_ + h) * DV_ + d) * S_ + s] = v;
      } else {
        C[(long)gm * ldc + gc] = v;
      }
    }
  }
}

// ---------------------------------------------------------------------------
// Fused attention: one wave (block=32) per (b, h, 16 q-rows).
//   scores (WMMA f32) -> scale/mask -> softmax (LDS row-block, 2-pass)
//   -> normalized attn streamed to global -> context = p@V (WMMA) -> ctxbuf
// Vt is the transposed V projection [B,H,DV,S] for contiguous B-fragments.
// ---------------------------------------------------------------------------
__global__ __launch_bounds__(32) void attn_kernel(
    const float* __restrict__ Qp, const float* __restrict__ Kp,
    const float* __restrict__ Vt, const unsigned char* __restrict__ mask,
    float* __restrict__ attn_g, float* __restrict__ ctxbuf)
{
  __shared__ float p[16 * S_];     // 32 KB unnormalized exp scores
  __shared__ float sinv[16];       // per-row 1/l

  const int lane = threadIdx.x & 31;
  const int half = lane >> 4;
  const int l16  = lane & 15;
  const int bid  = blockIdx.x;          // B*H*(S/16) = 4096
  const int qt   = bid & 31;
  const int h    = (bid >> 5) & 7;
  const int b    = bid >> 8;
  const int q0   = qt * 16;
  const long base = (long)(b * H_ + h) * S_;   // row base in [B*H, S, *]
  const float scale = 0.17677669529663687f;    // 1/sqrt(32)

  // Q A-fragments for full DK=32
  v2f aq[8];
  #pragma unroll
  for (int ks = 0; ks < 8; ++ks) {
    int kb = ks * 4 + 2 * half;
    aq[ks] = *(const v2f*)&Qp[(base + q0 + l16) * DK_ + kb];
  }

  // ---- scores -> masked/scaled -> LDS ----
  for (int kt = 0; kt < 32; ++kt) {
    v8f c = {};
    #pragma unroll
    for (int ks = 0; ks < 8; ++ks) {
      int kb = ks * 4 + 2 * half;
      v2f bv = *(const v2f*)&Kp[(base + kt * 16 + l16) * DK_ + kb];
      c = wmma4(aq[ks], bv, c);
    }
    #pragma unroll
    for (int r = 0; r < 8; ++r) {
      int row = r + 8 * half;
      int col = kt * 16 + l16;
      unsigned char mk = mask[(base + q0 + row) * S_ + col];
      p[row * S_ + col] = mk ? -1e9f : c[r] * scale;
    }
  }
  __syncthreads();

  // ---- softmax: lane covers row l16, columns half*256..+255 ----
  const int cb = half * 256;
  float m = -3.0e38f;
  for (int i = 0; i < 64; ++i) {
    float4 v = *(float4*)&p[l16 * S_ + cb + i * 4];
    m = fmaxf(m, fmaxf(fmaxf(v.x, v.y), fmaxf(v.z, v.w)));
  }
  m = fmaxf(m, __shfl_xor(m, 16, 32));
  float sum = 0.f;
  for (int i = 0; i < 64; ++i) {
    float4 v = *(float4*)&p[l16 * S_ + cb + i * 4];
    v.x = __expf(v.x - m); v.y = __expf(v.y - m);
    v.z = __expf(v.z - m); v.w = __expf(v.w - m);
    sum += v.x + v.y + v.z + v.w;
    *(float4*)&p[l16 * S_ + cb + i * 4] = v;
  }
  sum += __shfl_xor(sum, 16, 32);
  float invl = 1.0f / sum;
  if (half == 0) sinv[l16] = invl;
  __syncthreads();

  // ---- stream normalized attn tile to global (coalesced float4) ----
  float* ag = attn_g + (base + q0) * S_;
  for (int i = 0; i < 64; ++i) {
    int flat = (lane + 32 * i) * 4;          // 16*512 floats
    int row  = flat >> 9, col = flat & 511;
    float il = sinv[row];
    float4 v = *(float4*)&p[row * S_ + col];
    v.x *= il; v.y *= il; v.z *= il; v.w *= il;
    *(float4*)&ag[row * S_ + col] = v;
  }

  // ---- context = p @ V (16x32), contraction over 512 keys ----
  const float* vb = Vt + (long)(b * H_ + h) * DV_ * S_;  // [32 d][512 s]
  v8f c0 = {}; v8f c1 = {};
  for (int kt = 0; kt < 32; ++kt) {
    #pragma unroll
    for (int ks = 0; ks < 4; ++ks) {
      int kb = ks * 4 + 2 * half;
      v2f a  = *(const v2f*)&p[l16 * S_ + kt * 16 + kb];
      v2f b0 = *(const v2f*)&vb[(long)l16 * S_ + kt * 16 + kb];         // d=l16
      v2f b1 = *(const v2f*)&vb[(long)(16 + l16) * S_ + kt * 16 + kb];  // d=16+l16
      c0 = wmma4(a, b0, c0);
      c1 = wmma4(a, b1, c1);
    }
  }

  // ---- normalize + store into ctxbuf[B*S, 288] at cols h*32.. ----
  float* cbp = ctxbuf + ((long)b * S_ + q0) * (DV_ * (H_ + 1)) + h * DV_;
  #pragma unroll
  for (int r = 0; r < 8; ++r) {
    int row = r + 8 * half;
    float il = sinv[row];
    cbp[(long)row * 288 + l16]      = c0[r] * il;
    cbp[(long)row * 288 + 16 + l16] = c1[r] * il;
  }
}

// ---------------------------------------------------------------------------
// Gate: matrix_out[b,i,j] = matrix[b,i,j] *
//   sigmoid(MLP(LN([matrix[b,j,i], attn[b,0..7,j,i]])))
// 32x32 tile per block, LDS transpose so both gather and write are coalesced.
// ---------------------------------------------------------------------------
__global__ __launch_bounds__(256) void gate_kernel(
    const float* __restrict__ matrix, const float* __restrict__ attn_g,
    const float* __restrict__ fu_ln_g, const float* __restrict__ fu_ln_b,
    const float* __restrict__ fu_w1, const float* __restrict__ fu_b1,
    const float* __restrict__ fu_w2, const float* __restrict__ fu_b2,
    float* __restrict__ out_mat)
{
  __shared__ float gs[32][33];
  __shared__ float prm[85];     // 9 ln_g, 9 ln_b, 54 w1, 6 b1, 6 w2, 1 b2
  const int t = threadIdx.x;
  if (t < 85) {
    float v;
    if      (t <  9) v = fu_ln_g[t];
    else if (t < 18) v = fu_ln_b[t - 9];
    else if (t < 72) v = fu_w1[t - 18];
    else if (t < 78) v = fu_b1[t - 72];
    else if (t < 84) v = fu_w2[t - 78];
    else             v = fu_b2[0];
    prm[t] = v;
  }
  __syncthreads();

  const int b  = blockIdx.z;
  const int ti = blockIdx.y;   // i tile
  const int tj = blockIdx.x;   // j tile
  const long mbase = (long)b * S_ * S_;
  const long abase = (long)b * H_ * S_ * S_;

  #pragma unroll
  for (int ph = 0; ph < 4; ++ph) {
    int jj = (t >> 5) + 8 * ph;
    int ii = t & 31;
    int j = tj * 32 + jj, i = ti * 32 + ii;
    float v[9];
    v[0] = matrix[mbase + (long)j * S_ + i];           // coalesced over i
    #pragma unroll
    for (int hh = 0; hh < H_; ++hh)
      v[1 + hh] = attn_g[abase + ((long)hh * S_ + j) * S_ + i];
    float mean = 0.f;
    #pragma unroll
    for (int n = 0; n < 9; ++n) mean += v[n];
    mean *= (1.0f / 9.0f);
    float var = 0.f;
    #pragma unroll
    for (int n = 0; n < 9; ++n) { float d = v[n] - mean; var += d * d; }
    var *= (1.0f / 9.0f);
    float inv = rsqrtf(var + EPS_);
    float nv[9];
    #pragma unroll
    for (int n = 0; n < 9; ++n)
      nv[n] = prm[n] * ((v[n] - mean) * inv) + prm[9 + n];
    float z = prm[84];
    #pragma unroll
    for (int c = 0; c < 6; ++c) {
      float hs = prm[72 + c];
      #pragma unroll
      for (int n = 0; n < 9; ++n) hs += nv[n] * prm[18 + n * 6 + c];
      hs = fmaxf(hs, 0.f);
      z += hs * prm[78 + c];
    }
    gs[jj][ii] = 1.0f / (1.0f + __expf(-z));
  }
  __syncthreads();

  #pragma unroll
  for (int ph = 0; ph < 4; ++ph) {
    int il = (t >> 5) + 8 * ph;
    int jl = t & 31;
    int i = ti * 32 + il, j = tj * 32 + jl;
    long idx = mbase + (long)i * S_ + j;
    out_mat[idx] = matrix[idx] * gs[jl][il];           // coalesced over j
  }
}

// ---------------------------------------------------------------------------
// Row LayerNorm over 256 cols: one wave per row, 8 rows per 256-thread block.
// ---------------------------------------------------------------------------
__global__ __launch_bounds__(256) void ln_kernel(
    const float* __restrict__ x, const float* __restrict__ g,
    const float* __restrict__ bta, float* __restrict__ out)
{
  const int wv = threadIdx.x >> 5;
  const int lane = threadIdx.x & 31;
  const long row = (long)blockIdx.x * 8 + wv;
  const float* xr = x + row * DM_;
  float4 a = *(const float4*)&xr[lane * 4];
  float4 c = *(const float4*)&xr[128 + lane * 4];
  float s = a.x + a.y + a.z + a.w + c.x + c.y + c.z + c.w;
  #pragma unroll
  for (int m = 16; m >= 1; m >>= 1) s += __shfl_xor(s, m, 32);
  float mean = s * (1.0f / 256.0f);
  float vs = (a.x-mean)*(a.x-mean) + (a.y-mean)*(a.y-mean)
           + (a.z-mean)*(a.z-mean) + (a.w-mean)*(a.w-mean)
           + (c.x-mean)*(c.x-mean) + (c.y-mean)*(c.y-mean)
           + (c.z-mean)*(c.z-mean) + (c.w-mean)*(c.w-mean);
  #pragma unroll
  for (int m = 16; m >= 1; m >>= 1) vs += __shfl_xor(vs, m, 32);
  float inv = rsqrtf(vs * (1.0f / 256.0f) + EPS_);
  float4 g0 = *(const float4*)&g[lane * 4];
  float4 g1 = *(const float4*)&g[128 + lane * 4];
  float4 b0 = *(const float4*)&bta[lane * 4];
  float4 b1 = *(const float4*)&bta[128 + lane * 4];
  float4 o0, o1;
  o0.x = g0.x*(a.x-mean)*inv + b0.x;  o0.y = g0.y*(a.y-mean)*inv + b0.y;
  o0.z = g0.z*(a.z-mean)*inv + b0.z;  o0.w = g0.w*(a.w-mean)*inv + b0.w;
  o1.x = g1.x*(c.x-mean)*inv + b1.x;  o1.y = g1.y*(c.y-mean)*inv + b1.y;
  o1.z = g1.z*(c.z-mean)*inv + b1.z;  o1.w = g1.w*(c.w-mean)*inv + b1.w;
  *(float4*)&out[row * DM_ + lane * 4]       = o0;
  *(float4*)&out[row * DM_ + 128 + lane * 4] = o1;
}

// ---------------------------------------------------------------------------
extern "C" void kernel_launch(void* const* d_in, const int* in_sizes, int n_in,
                              void* d_out, int out_size, void* d_ws, size_t ws_size,
                              hipStream_t stream) {
  const float* inQ    = (const float*)d_in[0];
  const float* inK    = (const float*)d_in[1];
  const float* inV    = (const float*)d_in[2];
  const unsigned char* mask = (const unsigned char*)d_in[3];  // bool
  const float* matrix = (const float*)d_in[4];
  const float* W_Q    = (const float*)d_in[5];
  const float* W_K    = (const float*)d_in[6];
  const float* W_V    = (const float*)d_in[7];
  const float* W_V2   = (const float*)d_in[8];
  const float* W_fc   = (const float*)d_in[9];
  const float* ln_g   = (const float*)d_in[10];
  const float* ln_b   = (const float*)d_in[11];
  const float* fu_ln_g = (const float*)d_in[12];
  const float* fu_ln_b = (const float*)d_in[13];
  const float* fu_w1   = (const float*)d_in[14];
  const float* fu_b1   = (const float*)d_in[15];
  const float* fu_w2   = (const float*)d_in[16];
  const float* fu_b2   = (const float*)d_in[17];

  float* out    = (float*)d_out;                    // [B,S,256]
  float* outMat = out + (long)B_ * S_ * DM_;        // [B,S,S]

  // Workspace partition (floats)
  float* w      = (float*)d_ws;
  float* Qp     = w;                                // [B,H,S,32]   8 MB
  float* Kp     = Qp  + (long)B_ * H_ * S_ * DK_;   // [B,H,S,32]   8 MB
  float* Vt     = Kp  + (long)B_ * H_ * S_ * DK_;   // [B,H,32,S]   8 MB (transposed)
  float* V2p    = Vt  + (long)B_ * H_ * S_ * DV_;   // [B,S,32]     1 MB
  float* attnG  = V2p + (long)B_ * S_ * DV_;        // [B,H,S,S]  128 MB
  float* ctxbuf = attnG + (long)B_ * H_ * S_ * S_;  // [B,S,288]    9 MB
  float* fcout  = ctxbuf + (long)B_ * S_ * (DV_ * (H_ + 1));  // [B,S,256] 8 MB

  const int MBS = B_ * S_;   // 8192

  // 1-3: Q/K/V projections (Q,K head-split; V head-split transposed)
  gemm_wmma_f32<<<dim3(MBS/64, DM_/32, 1), 128, 0, stream>>>(
      inQ, W_Q, Qp, DM_, DM_, DM_, 0, 0, 0, 0, 1);
  gemm_wmma_f32<<<dim3(MBS/64, DM_/32, 1), 128, 0, stream>>>(
      inK, W_K, Kp, DM_, DM_, DM_, 0, 0, 0, 0, 1);
  gemm_wmma_f32<<<dim3(MBS/64, DM_/32, 1), 128, 0, stream>>>(
      inV, W_V, Vt, DM_, DM_, DM_, 0, 0, 0, 0, 2);
  // 4: V2p = input_V @ W_V2  (plain [B*S,32])
  gemm_wmma_f32<<<dim3(MBS/64, 1, 1), 128, 0, stream>>>(
      inV, W_V2, V2p, DM_, DM_, DV_, DV_, 0, 0, 0, 0);
  // 5: fused attention (scores+softmax+attn store+context)
  attn_kernel<<<dim3(B_ * H_ * (S_/16)), 32, 0, stream>>>(
      Qp, Kp, Vt, mask, attnG, ctxbuf);
  // 6: context2 = matrix @ V2p  -> ctxbuf cols 256..287 (batched)
  gemm_wmma_f32<<<dim3(S_/64, 1, B_), 128, 0, stream>>>(
      matrix, V2p, ctxbuf + H_ * DV_, S_, S_, DV_, 288,
      (long)S_ * S_, (long)S_ * DV_, (long)S_ * 288, 0);
  // 7: gate -> matrix_out
  gate_kernel<<<dim3(S_/32, S_/32, B_), 256, 0, stream>>>(
      matrix, attnG, fu_ln_g, fu_ln_b, fu_w1, fu_b1, fu_w2, fu_b2, outMat);
  // 8: fc = ctxbuf @ W_fc
  gemm_wmma_f32<<<dim3(MBS/64, DM_/32, 1), 128, 0, stream>>>(
      ctxbuf, W_fc, fcout, 288, 288, DM_, DM_, 0, 0, 0, 0);
  // 9: LayerNorm -> d_out
  ln_kernel<<<dim3(MBS/8), 256, 0, stream>>>(fcout, ln_g, ln_b, out);
}